// MaskDyanEncoder_14328010900068
// MI455X (gfx1250) — compile-verified
//
#include <hip/hip_runtime.h>
#include <math.h>

// ---------------------------------------------------------------------------
// MaskDyanEncoder (masked reweighted FISTA sparse coding) for gfx1250.
//
// 800 independent problems (B=16 x N=50). Per problem:
//   DM   = dic * mask          (36 x 161)
//   DtD  = DM^T DM             (161x161)  -> f32 WMMA 16x16x4, tiles in LDS
//   L    = lambda_max(DtD)     -> power iteration on LDS-resident DtD
//   FISTA: 100 iters of x <- shrink((I - DtD/L) y + DtY/L -+ w*lam/L)
//   two reweighting passes; global weight-norm reduction between passes.
// ---------------------------------------------------------------------------

typedef __attribute__((ext_vector_type(2))) float v2f;
typedef __attribute__((ext_vector_type(8))) float v8f;

namespace {
constexpr int CB = 16;          // batch
constexpr int CT = 36;          // sequence length
constexpr int CN = 50;          // joints
constexpr int CP = 80;          // pole pairs
constexpr int CK = 1 + 2 * CP;  // 161 dictionary atoms
constexpr int KP = 176;         // K padded to 11*16 for WMMA tiling
constexpr int KS = 177;         // LDS row stride (odd -> conflict-free)
constexpr int NPROB = CB * CN;  // 800
constexpr int NTILE = KP / 16;  // 11
constexpr int NTILES = NTILE * NTILE;  // 121
constexpr float LAM = 0.1f;
constexpr int MAXITER = 100;
constexpr float EPS = 1e-5f;

struct Smem {
  float DM[CT][KS];    // masked dictionary (padded cols are zero)
  float DtD[KP][KS];   // Gram matrix
  float mask[CT];
  float ym[CT];        // masked signal y[t] = m[t]*x[t]
  float DtY[KP];
  float vec[KP];       // power-iteration vector
  float xold[KP];
  float yv[KP];        // FISTA momentum vector
  float cminus[KP];
  float cplus[KP];
  float red[256];      // block reduction scratch
};
}  // namespace

__device__ __forceinline__ float block_reduce_sum(float v, float* red, int tid) {
  red[tid] = v;
  __syncthreads();
#pragma unroll
  for (int off = 128; off > 0; off >>= 1) {
    if (tid < off) red[tid] += red[tid + off];
    __syncthreads();
  }
  float r = red[0];
  __syncthreads();  // make red reusable immediately
  return r;
}

// ---------------------------------------------------------------------------
// Kernel 0: build the dictionary [T, KP] (zero-padded cols) + emit dic output.
// ---------------------------------------------------------------------------
__global__ void dict_kernel(const float* __restrict__ rr,
                            const float* __restrict__ theta,
                            float* __restrict__ dic_ws,
                            float* __restrict__ out_dic) {
  int idx = blockIdx.x * blockDim.x + threadIdx.x;
  if (idx >= CT * KP) return;
  int t = idx / KP, k = idx - t * KP;
  float ft = (float)t;
  float v = 0.0f;
  if (k == 0) {
    v = 1.0f;
  } else if (k <= CP) {
    v = powf(rr[k - 1], ft) * cosf(ft * theta[k - 1]);
  } else if (k < CK) {
    v = powf(rr[k - 1 - CP], ft) * sinf(ft * theta[k - 1 - CP]);
  }
  dic_ws[idx] = v;
  if (k < CK) out_dic[t * CK + k] = v;
}

// ---------------------------------------------------------------------------
// Kernel 1/3: one workgroup (8 wave32s) per problem.
// pass==0: FISTA with w=1, write temp + per-problem sum(w_raw^2).
// pass==1: FISTA with globally normalized reweighting, write outputs.
// ---------------------------------------------------------------------------
__global__ __launch_bounds__(256) void solve_kernel(
    const float* __restrict__ xin, const float* __restrict__ Min,
    const float* __restrict__ dic, int pass, float* __restrict__ temp_ws,
    float* __restrict__ wsum_partial, const float* __restrict__ wnorm_total,
    float* __restrict__ out_sc, float* __restrict__ out_rec) {
  extern __shared__ char smem_raw[];
  Smem& s = *reinterpret_cast<Smem*>(smem_raw);
  const int tid = threadIdx.x;
  const int prob = blockIdx.x;
  const int b = prob / CN;
  const int n = prob - b * CN;

  if (tid == 0) __builtin_prefetch(dic, 0, 3);  // global_prefetch_b8 (L2-hot dict)

  // ---- load mask + masked signal -----------------------------------------
  if (tid < CT) {
    float m = Min[(b * CT + tid) * CN + n];
    float xv = xin[(b * CT + tid) * CN + n];
    s.mask[tid] = m;
    s.ym[tid] = m * xv;
  }
  __syncthreads();

  // ---- masked dictionary into LDS ----------------------------------------
  for (int idx = tid; idx < CT * KP; idx += 256) {
    int t = idx / KP, k = idx - t * KP;
    s.DM[t][k] = dic[t * KP + k] * s.mask[t];
  }
  __syncthreads();

  // ---- DtD = DM^T * DM via V_WMMA_F32_16X16X4_F32 ------------------------
  // A[M=row][K=t] = DM[t][row]; B[K=t][N=col] = DM[t][col].
  // f32 16x4 A layout: lanes 0-15 hold K={k0,k0+1}, lanes 16-31 K={k0+2,k0+3}.
  const int wave = tid >> 5;
  const int lane = tid & 31;
  const int l15 = lane & 15;
  const int hi = lane >> 4;
  for (int tile = wave; tile < NTILES; tile += 8) {
    int ti = tile / NTILE, tj = tile - ti * NTILE;
    int row = ti * 16 + l15;
    int col = tj * 16 + l15;
    v8f acc = {0.f, 0.f, 0.f, 0.f, 0.f, 0.f, 0.f, 0.f};
#pragma unroll
    for (int kk = 0; kk < CT / 4; ++kk) {
      int k0 = kk * 4 + hi * 2;
      v2f a, bm;
      a.x = s.DM[k0][row];
      a.y = s.DM[k0 + 1][row];
      bm.x = s.DM[k0][col];
      bm.y = s.DM[k0 + 1][col];
      acc = __builtin_amdgcn_wmma_f32_16x16x4_f32(false, a, false, bm,
                                                  (short)0, acc, false, false);
    }
#pragma unroll
    for (int j = 0; j < 8; ++j)
      s.DtD[ti * 16 + hi * 8 + j][tj * 16 + l15] = acc[j];
  }

  // ---- DtY and power-iteration init (runs while WMMA tiles drain) --------
  if (tid < KP) {
    float acc = 0.f;
#pragma unroll
    for (int t = 0; t < CT; ++t) acc += s.DM[t][tid] * s.ym[t];
    s.DtY[tid] = acc;                       // zero for padded atoms
    s.vec[tid] = (tid < CK) ? 1.0f : 0.0f;  // power-iteration start vector
  }
  __syncthreads();

  // ---- L = lambda_max(DtD) by power iteration (PSD => max|eig|) ----------
  float L = 1.0f;
  for (int it = 0; it < 32; ++it) {
    float u = 0.f;
    if (tid < CK) {
      for (int j = 0; j < CK; ++j) u += s.DtD[tid][j] * s.vec[j];
    }
    float nrm2 = block_reduce_sum(u * u, s.red, tid);
    L = fmaxf(sqrtf(nrm2), 1e-20f);
    if (tid < CK) s.vec[tid] = u / L;
    __syncthreads();
  }
  const float Linv = 1.0f / L;

  // ---- per-atom reweighting weight ---------------------------------------
  float wk = 1.0f;
  if (pass == 1 && tid < CK) {
    float tprev = temp_ws[prob * CK + tid];
    float wr = 1.0f / (fabsf(tprev) + 0.01f);           // Wc==1
    float scale = ((float)(CK * CT)) / sqrtf(*wnorm_total);  // * K*T / ||w||
    wk = wr * scale;
  }

  if (tid < KP) {
    float c0 = (tid < CK) ? s.DtY[tid] * Linv : 0.f;
    float wl = (tid < CK) ? wk * LAM * Linv : 0.f;
    s.cminus[tid] = c0 - wl;
    s.cplus[tid] = c0 + wl;
    s.xold[tid] = 0.f;
    s.yv[tid] = 0.f;
  }
  __syncthreads();

  // ---- FISTA main loop: Ay = y - Linv*(DtD y) ----------------------------
  float tval = 1.0f;
  for (int iter = 0; iter < MAXITER; ++iter) {
    float xo = 0.f, xn = 0.f;
    if (tid < CK) {
      float acc = 0.f;
      for (int j = 0; j < CK; ++j) acc += s.DtD[tid][j] * s.yv[j];
      float Ay = s.yv[tid] - Linv * acc;
      xo = s.xold[tid];
      xn = fmaxf(Ay + s.cminus[tid], 0.f) + fminf(Ay + s.cplus[tid], 0.f);
    }
    float d = xn - xo;
    float dn2 = block_reduce_sum(d * d, s.red, tid);  // also fences yv reads
    float tnew = 0.5f * (1.0f + sqrtf(1.0f + 4.0f * tval * tval));
    if (tid < CK) {
      s.yv[tid] = xn + ((tval - 1.0f) / tnew) * (xn - xo);
      s.xold[tid] = xn;
    }
    tval = tnew;
    __syncthreads();
    if (sqrtf(dn2) * (1.0f / (float)CN) < EPS) break;  // uniform exit
  }
  __syncthreads();

  // ---- epilogue ----------------------------------------------------------
  if (pass == 0) {
    float wr2 = 0.f;
    if (tid < CK) {
      float xv = s.xold[tid];
      temp_ws[prob * CK + tid] = xv;
      float wr = 1.0f / (fabsf(xv) + 0.01f);
      wr2 = wr * wr;
    }
    float tot = block_reduce_sum(wr2, s.red, tid);
    if (tid == 0) wsum_partial[prob] = tot;
  } else {
    if (tid < CK) out_sc[(b * CK + tid) * CN + n] = s.xold[tid];  // [B,K,N]
    if (tid < CT) {  // reconst = D @ x with UNMASKED dictionary
      float acc = 0.f;
      for (int k = 0; k < CK; ++k) acc += dic[tid * KP + k] * s.xold[k];
      out_rec[(b * CN + n) * CT + tid] = acc;  // [B,N,T,1]
    }
  }
}

// ---------------------------------------------------------------------------
// Kernel 2: global reduction of per-problem sum(w^2) -> ||w||^2.
// ---------------------------------------------------------------------------
__global__ __launch_bounds__(256) void reduce_kernel(
    const float* __restrict__ part, float* __restrict__ total) {
  __shared__ float red[256];
  int tid = threadIdx.x;
  float s = 0.f;
  for (int i = tid; i < NPROB; i += 256) s += part[i];
  red[tid] = s;
  __syncthreads();
#pragma unroll
  for (int off = 128; off > 0; off >>= 1) {
    if (tid < off) red[tid] += red[tid + off];
    __syncthreads();
  }
  if (tid == 0) total[0] = red[0];
}

// ---------------------------------------------------------------------------
extern "C" void kernel_launch(void* const* d_in, const int* in_sizes, int n_in,
                              void* d_out, int out_size, void* d_ws,
                              size_t ws_size, hipStream_t stream) {
  // setup_inputs order: x [B,T,N,1] f32, T (int scalar, ==36), M [B,T,N,1] f32,
  //                     rr [P] f32, theta [P] f32
  const float* x = (const float*)d_in[0];
  const float* M = (const float*)d_in[2];
  const float* rr = (const float*)d_in[3];
  const float* theta = (const float*)d_in[4];

  float* out = (float*)d_out;
  float* out_sc = out;                   // sparseCode [B,K,N]  = 128800
  float* out_dic = out + CB * CK * CN;   // dic        [T,K]    = 5796
  float* out_rec = out_dic + CT * CK;    // reconst    [B,N,T,1]= 28800

  float* ws = (float*)d_ws;
  float* dic_ws = ws;                     // CT*KP padded dictionary
  float* temp_ws = dic_ws + CT * KP;      // NPROB*CK pass-0 codes
  float* wsum = temp_ws + NPROB * CK;     // NPROB partial weight norms
  float* wnorm = wsum + NPROB;            // 1 scalar ||w||^2

  dict_kernel<<<(CT * KP + 255) / 256, 256, 0, stream>>>(rr, theta, dic_ws,
                                                         out_dic);

  size_t smem = sizeof(Smem);  // ~152 KB, fits 320 KB/WGP (2 WGs per WGP)
  solve_kernel<<<NPROB, 256, smem, stream>>>(x, M, dic_ws, 0, temp_ws, wsum,
                                             wnorm, out_sc, out_rec);
  reduce_kernel<<<1, 256, 0, stream>>>(wsum, wnorm);
  solve_kernel<<<NPROB, 256, smem, stream>>>(x, M, dic_ws, 1, temp_ws, wsum,
                                             wnorm, out_sc, out_rec);
}